// MoEFeedForward_4612794876260
// MI455X (gfx1250) — compile-verified
//
#include <hip/hip_runtime.h>

typedef __attribute__((ext_vector_type(2))) float v2f;
typedef __attribute__((ext_vector_type(8))) float v8f;

#define NTOK 2048
#define DDIM 1024
#define FDIM 4096
#define NEXP 8
#define RMS_EPS 1e-6f
#define MTILE 32  // rows (tokens) per block tile: 2 x 16-row WMMA subtiles

// ---------------------------------------------------------------------------
// Zero the per-expert token counters (must happen every call; graph-safe).
// ---------------------------------------------------------------------------
__global__ void moe_init(int* __restrict__ counts) {
  if (threadIdx.x < NEXP) counts[threadIdx.x] = 0;
}

// ---------------------------------------------------------------------------
// Per-token: RMSNorm, gate logits, softmax/top-1, append to expert list.
// One 256-thread block per token; D=1024 -> 4 floats per thread.
// ---------------------------------------------------------------------------
__global__ __launch_bounds__(256)
void moe_gate(const float* __restrict__ x, const float* __restrict__ rms_w,
              const float* __restrict__ gate_w,
              float* __restrict__ xn, float* __restrict__ score,
              int* __restrict__ tokmap, int* __restrict__ counts) {
  const int t = blockIdx.x;
  const int tid = threadIdx.x;
  __shared__ float red[256];
  __shared__ float lred[NEXP][256];

  const float4 v = ((const float4*)(x + (size_t)t * DDIM))[tid];
  red[tid] = v.x * v.x + v.y * v.y + v.z * v.z + v.w * v.w;
  __syncthreads();
  for (int s = 128; s > 0; s >>= 1) {
    if (tid < s) red[tid] += red[tid + s];
    __syncthreads();
  }
  const float inv = rsqrtf(red[0] * (1.0f / (float)DDIM) + RMS_EPS);

  const float4 g = ((const float4*)rms_w)[tid];
  float4 xv;
  xv.x = v.x * inv * g.x;
  xv.y = v.y * inv * g.y;
  xv.z = v.z * inv * g.z;
  xv.w = v.w * inv * g.w;
  ((float4*)(xn + (size_t)t * DDIM))[tid] = xv;

#pragma unroll
  for (int e = 0; e < NEXP; ++e) {
    const float4 gw = ((const float4*)(gate_w + (size_t)e * DDIM))[tid];
    lred[e][tid] = xv.x * gw.x + xv.y * gw.y + xv.z * gw.z + xv.w * gw.w;
  }
  __syncthreads();
  for (int s = 128; s > 0; s >>= 1) {
    if (tid < s) {
#pragma unroll
      for (int e = 0; e < NEXP; ++e) lred[e][tid] += lred[e][tid + s];
    }
    __syncthreads();
  }

  if (tid == 0) {
    float lmax = lred[0][0];
    int arg = 0;
#pragma unroll
    for (int e = 1; e < NEXP; ++e) {
      const float l = lred[e][0];
      if (l > lmax) { lmax = l; arg = e; }   // strict > == first-max (argmax tie rule)
    }
    float sum = 0.0f;
#pragma unroll
    for (int e = 0; e < NEXP; ++e) sum += __expf(lred[e][0] - lmax);
    score[t] = 1.0f / sum;                   // softmax value at the argmax
    const int pos = atomicAdd(&counts[arg], 1);
    tokmap[arg * NTOK + pos] = t;
  }
}

// ---------------------------------------------------------------------------
// Gathered-token GEMM using V_WMMA_F32_16X16X4_F32.
//   A: [NTOK, KDIM] token-indexed (gather via tokmap)   (xn or h)
//   W: [NEXP, KDIM, NDIM]                                (W1 or W2)
//   bias: [NEXP, NDIM]
//   SILU=true : Out[tok] = silu(A@W + b)      (h buffer)
//   SILU=false: Out[tok] = (A@W + b) * score  (final output)
// Block = 128 threads = 4 waves. Block tile: M=32, N=256.
// Per wave: 2 M-subtiles x 4 N-subtiles = 8 accumulators (64 VGPRs);
// each 256B B-fragment feeds 2 WMMAs. K staged in LDS in chunks of 64.
// ---------------------------------------------------------------------------
template <int KDIM, int NDIM, bool SILU>
__global__ __launch_bounds__(128)
void moe_gemm(const float* __restrict__ Aglob, const float* __restrict__ Wglob,
              const float* __restrict__ bias, const int* __restrict__ tokmap,
              const int* __restrict__ counts, const float* __restrict__ score,
              float* __restrict__ Out) {
  const int e = blockIdx.z;
  const int cnt = counts[e];
  const int m0 = blockIdx.y * MTILE;
  if (m0 >= cnt) return;  // uniform per block: full-EXEC waves proceed

  __shared__ int tokS[MTILE];
  __shared__ float As[MTILE][68];  // +4 pad: conflict-free strided reads

  const int tid = threadIdx.x;
  if (tid < MTILE)
    tokS[tid] = (m0 + tid < cnt) ? tokmap[e * NTOK + m0 + tid] : -1;
  __syncthreads();

  const float* __restrict__ We = Wglob + (size_t)e * KDIM * NDIM;
  const int lane = tid & 31;
  const int wave = tid >> 5;
  const int lr = lane & 15;           // M-row / N-col within 16
  const int kh2 = (lane >> 4) << 1;   // lanes 0-15 -> K0/K1, lanes 16-31 -> K2/K3
  const int nbase = blockIdx.x * 256 + wave * 64 + lr;

  v8f acc[2][4] = {};

  for (int kc = 0; kc < KDIM; kc += 64) {
    // Stage gathered A chunk [32 x 64] into LDS (zeros for padded rows).
#pragma unroll
    for (int i = tid; i < MTILE * 64; i += 128) {
      const int r = i >> 6;
      const int k = i & 63;
      const int tk = tokS[r];
      As[r][k] = (tk >= 0) ? Aglob[(size_t)tk * KDIM + kc + k] : 0.0f;
    }
    __syncthreads();

    if (kc + 64 < KDIM)  // pull next weight chunk toward L2 (global_prefetch_b8)
      __builtin_prefetch(We + (size_t)(kc + 64 + kh2) * NDIM + nbase, 0, 0);

#pragma unroll 4
    for (int kk = 0; kk < 16; ++kk) {
      // A fragments 16x4 f32: lane<16 holds (K=k0,k0+1), lane>=16 (K=k0+2,k0+3)
      v2f a0, a1;
      a0.x = As[lr][kk * 4 + kh2 + 0];
      a0.y = As[lr][kk * 4 + kh2 + 1];
      a1.x = As[lr + 16][kk * 4 + kh2 + 0];
      a1.y = As[lr + 16][kk * 4 + kh2 + 1];
      const float* Bp = We + (size_t)(kc + kk * 4 + kh2) * NDIM + nbase;
#pragma unroll
      for (int s = 0; s < 4; ++s) {
        // B fragment 4x16 f32: same K split across lane halves, N = lr
        v2f b;
        b.x = Bp[s * 16];
        b.y = Bp[(size_t)NDIM + s * 16];
        acc[0][s] = __builtin_amdgcn_wmma_f32_16x16x4_f32(
            false, a0, false, b, (short)0, acc[0][s], false, false);
        acc[1][s] = __builtin_amdgcn_wmma_f32_16x16x4_f32(
            false, a1, false, b, (short)0, acc[1][s], false, false);
      }
    }
    __syncthreads();
  }

  // C/D layout: VGPR r holds M=r (lanes 0-15) or M=r+8 (lanes 16-31); N = lr.
  const int mo = (lane < 16) ? 0 : 8;
#pragma unroll
  for (int m = 0; m < 2; ++m) {
#pragma unroll
    for (int r = 0; r < 8; ++r) {
      const int row = m * 16 + r + mo;
      const int tk = tokS[row];
      if (tk < 0) continue;
#pragma unroll
      for (int s = 0; s < 4; ++s) {
        const int col = nbase + s * 16;
        float val = acc[m][s][r] + bias[(size_t)e * NDIM + col];
        if (SILU) {
          val = val * (1.0f / (1.0f + __expf(-val)));          // silu
          Out[(size_t)tk * NDIM + col] = val;                  // h buffer
        } else {
          Out[(size_t)tk * NDIM + col] = val * score[tk];      // final out
        }
      }
    }
  }
}

// ---------------------------------------------------------------------------
extern "C" void kernel_launch(void* const* d_in, const int* in_sizes, int n_in,
                              void* d_out, int out_size, void* d_ws, size_t ws_size,
                              hipStream_t stream) {
  const float* x      = (const float*)d_in[0];
  const float* rms_w  = (const float*)d_in[1];
  const float* gate_w = (const float*)d_in[2];
  const float* W1     = (const float*)d_in[3];
  const float* b1     = (const float*)d_in[4];
  const float* W2     = (const float*)d_in[5];
  const float* b2     = (const float*)d_in[6];
  float* out = (float*)d_out;

  // Workspace layout (floats): xn | h | score | tokmap | counts
  float* ws     = (float*)d_ws;
  float* xn     = ws;                               // NTOK*DDIM
  float* hbuf   = xn + (size_t)NTOK * DDIM;         // NTOK*FDIM
  float* score  = hbuf + (size_t)NTOK * FDIM;       // NTOK
  int*   tokmap = (int*)(score + NTOK);             // NEXP*NTOK
  int*   counts = tokmap + (size_t)NEXP * NTOK;     // NEXP

  moe_init<<<1, 32, 0, stream>>>(counts);
  moe_gate<<<NTOK, 256, 0, stream>>>(x, rms_w, gate_w, xn, score, tokmap, counts);

  // GEMM1: [cnt_e x 1024] @ [1024 x 4096], fused bias+SiLU -> h (token-indexed)
  moe_gemm<DDIM, FDIM, true>
      <<<dim3(FDIM / 256, NTOK / MTILE, NEXP), 128, 0, stream>>>(
          xn, W1, b1, tokmap, counts, nullptr, hbuf);

  // GEMM2: [cnt_e x 4096] @ [4096 x 1024], fused bias + top1-score scale -> out
  moe_gemm<FDIM, DDIM, false>
      <<<dim3(DDIM / 256, NTOK / MTILE, NEXP), 128, 0, stream>>>(
          hbuf, W2, b2, tokmap, counts, score, out);
}